// GaussianKernelLayer_20057497272936
// MI455X (gfx1250) — compile-verified
//
#include <hip/hip_runtime.h>

typedef __attribute__((ext_vector_type(2))) float v2f;
typedef __attribute__((ext_vector_type(8))) float v8f;

#define N_PTS 500000
#define M_CTR 256
// 1 / ((2*pi)^(4/2) * 1^4) = 1/(4*pi^2)
#define NORM_CONST 0.025330295910584444f

// ---------------------------------------------------------------------------
// Preamble: w[m] = softmax(coef)[m] * NORM_CONST * exp(-0.5*|c_m|^2)
// One block of 256 threads (M_CTR == 256).
// ---------------------------------------------------------------------------
__global__ void gauss_weights_kernel(const float* __restrict__ centers,
                                     const float* __restrict__ coefficients,
                                     float* __restrict__ w) {
    __shared__ float red[M_CTR];
    const int t = threadIdx.x;
    float c = coefficients[t];

    // max-reduce for stable softmax
    red[t] = c;
    __syncthreads();
    for (int s = M_CTR / 2; s > 0; s >>= 1) {
        if (t < s) red[t] = fmaxf(red[t], red[t + s]);
        __syncthreads();
    }
    const float mx = red[0];
    __syncthreads();

    const float e = __expf(c - mx);
    red[t] = e;
    __syncthreads();
    for (int s = M_CTR / 2; s > 0; s >>= 1) {
        if (t < s) red[t] += red[t + s];
        __syncthreads();
    }
    const float sum = red[0];

    const float c0 = centers[t * 4 + 0];
    const float c1 = centers[t * 4 + 1];
    const float c2 = centers[t * 4 + 2];
    const float c3 = centers[t * 4 + 3];
    const float cn2 = c0 * c0 + c1 * c1 + c2 * c2 + c3 * c3;

    w[t] = (e / sum) * NORM_CONST * __expf(-0.5f * cn2);
}

// ---------------------------------------------------------------------------
// Main: one wave32 handles 16 x-rows against all 256 centers.
// dot = x @ centers.T via V_WMMA_F32_16X16X4_F32 (K = D = 4, exact fit).
// out[n] = exp(-0.5*|x_n|^2) * sum_m w[m] * exp(dot[n,m])
// ---------------------------------------------------------------------------
__global__ void __launch_bounds__(256) gauss_main_kernel(
        const float* __restrict__ x,
        const float* __restrict__ centers,
        const float* __restrict__ w,
        float* __restrict__ out) {
    __shared__ float ws[M_CTR];
    ws[threadIdx.x] = w[threadIdx.x];   // blockDim.x == 256
    __syncthreads();

    const int lane = threadIdx.x & 31;
    const int waveId = blockIdx.x * (blockDim.x >> 5) + (threadIdx.x >> 5);
    const int numTiles = N_PTS / 16;    // 31250, exact
    if (waveId >= numTiles) return;     // wave-uniform: EXEC stays all-ones

    const int row0 = waveId * 16;
    const int r    = lane & 15;         // row (for A) / column (for B,C,D)
    const int hi   = lane >> 4;         // half-wave: K-split {0,1} vs {2,3}
    const int koff = hi * 2;

    // A fragment: lane holds x[row0+r][koff], x[row0+r][koff+1]
    v2f a = *(const v2f*)(x + (size_t)(row0 + r) * 4 + koff);

    // per-row |x|^2: each half-wave has half the squares; combine across halves
    float part = a.x * a.x + a.y * a.y;
    float x2 = part + __shfl_xor(part, 16);   // lane L now holds |x_{row0+L%16}|^2

    float acc[8];
#pragma unroll
    for (int j = 0; j < 8; ++j) acc[j] = 0.0f;

#pragma unroll
    for (int tile = 0; tile < 16; ++tile) {
        const int col = tile * 16 + r;
        // B fragment: lane holds centers[col][koff], centers[col][koff+1]
        v2f b = *(const v2f*)(centers + col * 4 + koff);

        v8f c = {};
        // D = A(16x4) x B(4x16) + 0 : full dot products x_row . c_col
        c = __builtin_amdgcn_wmma_f32_16x16x4_f32(
                /*neg_a=*/false, a, /*neg_b=*/false, b,
                /*c_mod=*/(short)0, c, /*reuse_a=*/false, /*reuse_b=*/false);

        const float wcol = ws[col];
#pragma unroll
        for (int j = 0; j < 8; ++j)
            acc[j] += wcol * __expf(c[j]);
    }

    // Reduce over the 16 lanes of each half-wave (columns of the output tile).
    // Rows 0-7 live in lanes 0-15, rows 8-15 in lanes 16-31 (C/D layout).
#pragma unroll
    for (int j = 0; j < 8; ++j) {
        float v = acc[j];
        v += __shfl_xor(v, 1);
        v += __shfl_xor(v, 2);
        v += __shfl_xor(v, 4);
        v += __shfl_xor(v, 8);
        acc[j] = v;
    }

    // Apply per-row exp(-0.5*|x|^2); row for acc[j] is (hi*8 + j).
#pragma unroll
    for (int j = 0; j < 8; ++j) {
        const float x2j = __shfl(x2, hi * 8 + j);   // x2 of row hi*8+j (lanes j / 8+j)
        acc[j] *= __expf(-0.5f * x2j);
    }

    // Lane 0 stores rows 0-7, lane 16 stores rows 8-15 (contiguous).
    if (r == 0) {
        const int base = row0 + hi * 8;
#pragma unroll
        for (int j = 0; j < 8; ++j) out[base + j] = acc[j];
    }
}

// ---------------------------------------------------------------------------
extern "C" void kernel_launch(void* const* d_in, const int* in_sizes, int n_in,
                              void* d_out, int out_size, void* d_ws, size_t ws_size,
                              hipStream_t stream) {
    const float* x            = (const float*)d_in[0];   // (N, 4)
    const float* centers      = (const float*)d_in[1];   // (256, 4)
    const float* coefficients = (const float*)d_in[2];   // (256,)
    float* out = (float*)d_out;                          // (N,)
    float* w   = (float*)d_ws;                           // 256 floats scratch

    gauss_weights_kernel<<<1, M_CTR, 0, stream>>>(centers, coefficients, w);

    const int numWaveTiles = N_PTS / 16;                 // 31250
    const int wavesPerBlock = 8;                         // 256 threads
    const int blocks = (numWaveTiles + wavesPerBlock - 1) / wavesPerBlock;
    gauss_main_kernel<<<blocks, 256, 0, stream>>>(x, centers, w, out);
}